// mLSTMCell_35656818491883
// MI455X (gfx1250) — compile-verified
//
#include <hip/hip_runtime.h>
#include <hip/hip_bf16.h>

#define BB   4
#define SS   2048
#define DIMM 1024
#define NHH  8
#define DHH  128
#define CSZ  16
#define NSS  128

typedef __attribute__((ext_vector_type(2))) float v2f;
typedef __attribute__((ext_vector_type(8))) float v8f;

// D = A(16x4,f32) * B(4x16,f32) + C(16x16,f32) — native CDNA5 f32 matrix op
__device__ inline v8f wmma_f32_k4(v2f a, v2f b, v8f c) {
  return __builtin_amdgcn_wmma_f32_16x16x4_f32(false, a, false, b, (short)0, c, false, false);
}

// ---- fragment helpers (wave32 layouts per CDNA5 ISA) ----
// A 16x4: lanes 0-15 hold M=lane, K={k0,k0+1}; lanes 16-31 hold M=lane-16, K={k0+2,k0+3}
__device__ inline v2f frag_a_row(const float* m, int ld, int k0, int lane) {
  int r = lane & 15;
  int kb = k0 + ((lane >> 4) << 1);
  v2f a; a[0] = m[r * ld + kb]; a[1] = m[r * ld + kb + 1]; return a;
}
__device__ inline v2f frag_a_row_scaled(const float* m, int ld, int k0, int lane,
                                        const float* rowscale) {
  int r = lane & 15;
  int kb = k0 + ((lane >> 4) << 1);
  float s = rowscale[r];
  v2f a; a[0] = m[r * ld + kb] * s; a[1] = m[r * ld + kb + 1] * s; return a;
}
// A[r][k] = m[k][row0+r]  (transposed source, e.g. k^T)
__device__ inline v2f frag_a_transpose(const float* m, int ld, int row0, int k0, int lane) {
  int r = row0 + (lane & 15);
  int kb = k0 + ((lane >> 4) << 1);
  v2f a; a[0] = m[kb * ld + r]; a[1] = m[(kb + 1) * ld + r]; return a;
}
// B 4x16 row-striped: lane's column N=lane&15; lanes 0-15 K={k0,k0+1}, lanes 16-31 K={k0+2,k0+3}
__device__ inline v2f frag_b(const float* m, int ld, int k0, int col0, int lane) {
  int n = col0 + (lane & 15);
  int kb = k0 + ((lane >> 4) << 1);
  v2f b; b[0] = m[kb * ld + n]; b[1] = m[(kb + 1) * ld + n]; return b;
}
// B[k][n] = m[col0+n][k]  (transposed source, e.g. q·k^T needs k^T as B)
__device__ inline v2f frag_b_transpose(const float* m, int ld, int k0, int col0, int lane) {
  int n = col0 + (lane & 15);
  int kb = k0 + ((lane >> 4) << 1);
  v2f b; b[0] = m[n * ld + kb]; b[1] = m[n * ld + kb + 1]; return b;
}
// C/D 16x16 f32: VGPR r -> lanes 0-15: M=r, N=lane; lanes 16-31: M=r+8, N=lane-16
__device__ inline v8f frag_c_load(const float* m, int ld, int col0, int lane) {
  int n = col0 + (lane & 15);
  int rb = (lane >> 4) << 3;
  v8f c;
#pragma unroll
  for (int r = 0; r < 8; r++) c[r] = m[(rb + r) * ld + n];
  return c;
}
__device__ inline void frag_d_store(float* m, int ld, int col0, v8f d, int lane) {
  int n = col0 + (lane & 15);
  int rb = (lane >> 4) << 3;
#pragma unroll
  for (int r = 0; r < 8; r++) m[(rb + r) * ld + n] = d[r];
}

// =====================================================================
// Kernel 1: gate projection as a GEMM with N=16 (8 ig heads | 8 fg heads)
// 8 waves per 16-position block; wave w owns the interleaved K-slice
// {w*4 + 32*i} so q/k/v boundaries never split a K-step: three
// branch-free K loops. Partial tiles summed serially through LDS.
// =====================================================================
__global__ __launch_bounds__(256) void mlstm_gate_kernel(
    const float* __restrict__ q, const float* __restrict__ k, const float* __restrict__ v,
    const float* __restrict__ Wi, const float* __restrict__ bi,
    const float* __restrict__ Wf, const float* __restrict__ bf,
    float* __restrict__ igw, float* __restrict__ lfw) {
  __shared__ float part[8 * CSZ * CSZ];

  const int blk = blockIdx.x;          // 0 .. B*NS-1
  const int b = blk / NSS;
  const int s0 = (blk % NSS) * CSZ;
  const int tid = threadIdx.x;
  const int w = tid >> 5;
  const int ln = tid & 31;

  const int m = ln & 15;               // A row (seq position) == B col (head/gate)
  const int koff = (ln >> 4) << 1;     // 0 or 2
  const float* wrow = (m < 8) ? (Wi + (size_t)m * 3 * DIMM)
                              : (Wf + (size_t)(m - 8) * 3 * DIMM);
  const float* qrow = q + (size_t)(b * SS + s0 + m) * DIMM;
  const float* krow = k + (size_t)(b * SS + s0 + m) * DIMM;
  const float* vrow = v + (size_t)(b * SS + s0 + m) * DIMM;
  const int kbase = w * 4 + koff;      // 8-byte aligned (even element index)

  v8f acc = {0.f, 0.f, 0.f, 0.f, 0.f, 0.f, 0.f, 0.f};
#pragma unroll 4
  for (int i = 0; i < 32; i++) {       // q segment: K = [0, 1024)
    int kb = kbase + i * 32;
    v2f a = *(const v2f*)(qrow + kb);
    v2f bb = *(const v2f*)(wrow + kb);
    acc = wmma_f32_k4(a, bb, acc);
  }
#pragma unroll 4
  for (int i = 0; i < 32; i++) {       // k segment: K = [1024, 2048)
    int kb = kbase + i * 32;
    v2f a = *(const v2f*)(krow + kb);
    v2f bb = *(const v2f*)(wrow + DIMM + kb);
    acc = wmma_f32_k4(a, bb, acc);
  }
#pragma unroll 4
  for (int i = 0; i < 32; i++) {       // v segment: K = [2048, 3072)
    int kb = kbase + i * 32;
    v2f a = *(const v2f*)(vrow + kb);
    v2f bb = *(const v2f*)(wrow + 2 * DIMM + kb);
    acc = wmma_f32_k4(a, bb, acc);
  }
  frag_d_store(part + w * (CSZ * CSZ), CSZ, 0, acc, ln);
  __syncthreads();

  // one thread per (position c, column j) of the 16x16 gate tile
  const int c = tid >> 4;
  const int j = tid & 15;
  float g = 0.f;
#pragma unroll
  for (int wv = 0; wv < 8; wv++) g += part[wv * (CSZ * CSZ) + tid];
  const int s = s0 + c;
  if (j < 8) {
    igw[(size_t)(b * NHH + j) * SS + s] = g + bi[j];
  } else {
    float x = g + bf[j - 8];
    float lf = fminf(x, 0.f) - log1pf(expf(-fabsf(x)));   // stable log-sigmoid
    lfw[(size_t)(b * NHH + (j - 8)) * SS + s] = lf;
  }
}

// =====================================================================
// Kernel 2: fused chunkwise scan. One workgroup per (b,h); carry state
// C (128x128), n (128), m kept in LDS across the sequential chunk loop.
// 8 waves each own a 16-wide e-slice of every 16x128 GEMM.
// =====================================================================
__global__ __launch_bounds__(256) void mlstm_chunk_kernel(
    const float* __restrict__ q, const float* __restrict__ k, const float* __restrict__ v,
    const float* __restrict__ igw, const float* __restrict__ lfw,
    const float* __restrict__ lnw, const float* __restrict__ lnb,
    float* __restrict__ out) {
  __shared__ float Cs[DHH * DHH];                    // running C state (64 KB)
  __shared__ float qs[CSZ * DHH], ks[CSZ * DHH], vs[CSZ * DHH];
  __shared__ float vg[CSZ * DHH], hs[CSZ * DHH];
  __shared__ float Es[CSZ * CSZ];
  __shared__ float ns[DHH];
  __shared__ float igc[CSZ], lfc[CSZ], lfa[CSZ], lgs[CSZ], gex[CSZ];
  __shared__ float stb[CSZ], dcy[CSZ], nrm[CSZ], mu[CSZ], rsd[CSZ];
  __shared__ float m_state, m_new_s, a_s, b_s, m_loc_s;

  const int tid = threadIdx.x;
  const int w = tid >> 5;
  const int ln = tid & 31;
  const int bh = blockIdx.x;
  const int b = bh / NHH;
  const int h = bh % NHH;

  for (int i = tid; i < DHH * DHH; i += 256) Cs[i] = 0.f;
  if (tid < DHH) ns[tid] = 0.f;
  if (tid == 0) m_state = 0.f;
  __syncthreads();

  const float qscale = 0.08838834764831845f;  // 1/sqrt(DH)
  const float* igp = igw + (size_t)(b * NHH + h) * SS;
  const float* lfp = lfw + (size_t)(b * NHH + h) * SS;

  for (int nch = 0; nch < NSS; ++nch) {
    const int s0 = nch * CSZ;

    // ---- stage tiles into LDS ----
    for (int i = tid; i < CSZ * DHH; i += 256) {
      int c = i >> 7, d = i & (DHH - 1);
      size_t g = (size_t)(b * SS + s0 + c) * DIMM + h * DHH + d;
      qs[i] = q[g] * qscale;
      ks[i] = k[g];
      vs[i] = v[g];
    }
    if (tid < CSZ) { igc[tid] = igp[s0 + tid]; lfc[tid] = lfp[s0 + tid]; }
    // prefetch next chunk (one lane per 128B line) -> global_prefetch_b8
    if (nch + 1 < NSS && tid < 64) {
      int c = tid >> 2, d = (tid & 3) * 32;
      size_t g = (size_t)(b * SS + s0 + CSZ + c) * DIMM + h * DHH + d;
      __builtin_prefetch(&q[g], 0, 1);
      __builtin_prefetch(&k[g], 0, 1);
      __builtin_prefetch(&v[g], 0, 1);
    }
    __syncthreads();

    // ---- chunk-level scalars ----
    if (tid == 0) {
      float acc = 0.f;
      for (int c = 0; c < CSZ; c++) { acc += lfc[c]; lfa[c] = acc; }
      float lfl = lfa[CSZ - 1];
      float mloc = -3.0e38f;
      for (int c = 0; c < CSZ; c++) {
        float lg = igc[c] - lfa[c] + lfl;
        lgs[c] = lg;
        mloc = fmaxf(mloc, lg);
      }
      m_loc_s = mloc;
      float mn = fmaxf(lfl + m_state, mloc);
      m_new_s = mn;
      a_s = expf(lfl + m_state - mn);
      b_s = expf(mloc - mn);
    }
    __syncthreads();

    if (tid < CSZ) {
      gex[tid] = expf(lgs[tid] - m_loc_s);
      float dmax = -3.0e38f;
      for (int j = 0; j <= tid; j++) dmax = fmaxf(dmax, lfa[tid] - lfa[j] + igc[j]);
      float st = fmaxf(dmax, m_state + lfa[tid]);
      stb[tid] = st;
      dcy[tid] = expf(m_state + lfa[tid] - st);
    }
    __syncthreads();

    // ---- inter_C = (q*decay) @ C_prev   [16x128, K=128]
    //      wave 0 also: Sqk = q @ k^T     [16x16,  K=128]
    {
      const int e0 = w * 16;
      v8f acc = {0.f, 0.f, 0.f, 0.f, 0.f, 0.f, 0.f, 0.f};
#pragma unroll 4
      for (int k0 = 0; k0 < DHH; k0 += 4) {
        v2f a = frag_a_row_scaled(qs, DHH, k0, ln, dcy);
        v2f bb = frag_b(Cs, DHH, k0, e0, ln);
        acc = wmma_f32_k4(a, bb, acc);
      }
      frag_d_store(hs, DHH, e0, acc, ln);
      if (w == 0) {
        v8f a2 = {0.f, 0.f, 0.f, 0.f, 0.f, 0.f, 0.f, 0.f};
#pragma unroll 4
        for (int k0 = 0; k0 < DHH; k0 += 4) {
          v2f a = frag_a_row(qs, DHH, k0, ln);
          v2f bb = frag_b_transpose(ks, DHH, k0, 0, ln);
          a2 = wmma_f32_k4(a, bb, a2);
        }
        frag_d_store(Es, CSZ, 0, a2, ln);
      }
    }
    __syncthreads();

    // ---- E = Sqk * exp(logD - stab), causal mask ----
    {
      int c = tid >> 4, j = tid & 15;
      float e = 0.f;
      if (j <= c) e = Es[tid] * expf(lfa[c] - lfa[j] + igc[j] - stb[c]);
      Es[tid] = e;
    }
    __syncthreads();

    // ---- norm per row: max(|sum(E) + qd·n|, exp(-stab)) + eps ----
    if (tid < CSZ) {
      float se = 0.f;
      for (int j = 0; j < CSZ; j++) se += Es[tid * CSZ + j];
      float inn = 0.f, dc = dcy[tid];
      for (int d = 0; d < DHH; d++) inn += qs[tid * DHH + d] * dc * ns[d];
      nrm[tid] = fmaxf(fabsf(se + inn), expf(-stb[tid])) + 1e-6f;
    }
    __syncthreads();

    // ---- apply norm; build v*gexp ----
    for (int i = tid; i < CSZ * DHH; i += 256) {
      int c = i >> 7;
      hs[i] = hs[i] / nrm[c];
      vg[i] = vs[i] * gex[c];
    }
    Es[tid] = Es[tid] / nrm[tid >> 4];  // 256 threads == 16x16 elements
    __syncthreads();

    // ---- h = (E/norm) @ v + interC/norm  (interC preloaded into WMMA C) ----
    {
      const int e0 = w * 16;
      v8f c8 = frag_c_load(hs, DHH, e0, ln);
#pragma unroll
      for (int k0 = 0; k0 < CSZ; k0 += 4) {
        v2f a = frag_a_row(Es, CSZ, k0, ln);
        v2f bb = frag_b(vs, DHH, k0, e0, ln);
        c8 = wmma_f32_k4(a, bb, c8);
      }
      frag_d_store(hs, DHH, e0, c8, ln);
    }
    __syncthreads();

    // ---- per-head layernorm + affine, write output ----
    if (tid < CSZ) {
      float s = 0.f;
      for (int d = 0; d < DHH; d++) s += hs[tid * DHH + d];
      float mea = s / (float)DHH;
      float va = 0.f;
      for (int d = 0; d < DHH; d++) { float t = hs[tid * DHH + d] - mea; va += t * t; }
      mu[tid] = mea;
      rsd[tid] = rsqrtf(va / (float)DHH + 1e-5f);
    }
    __syncthreads();

    for (int i = tid; i < CSZ * DHH; i += 256) {
      int c = i >> 7, d = i & (DHH - 1);
      int gd = h * DHH + d;
      float xn = (hs[i] - mu[c]) * rsd[c];
      out[(size_t)(b * SS + s0 + c) * DIMM + gd] = xn * (1.f + lnw[gd]) + lnb[gd];
    }

    // ---- state update: C = C*a + (k^T @ (v*gexp))*b ; each wave owns 16 d-rows ----
    {
      const int d0 = w * 16;
      for (int et = 0; et < 8; ++et) {
        const int e0 = et * 16;
        v8f acc = {0.f, 0.f, 0.f, 0.f, 0.f, 0.f, 0.f, 0.f};
#pragma unroll
        for (int k0 = 0; k0 < CSZ; k0 += 4) {
          v2f a = frag_a_transpose(ks, DHH, d0, k0, ln);
          v2f bb = frag_b(vg, DHH, k0, e0, ln);
          acc = wmma_f32_k4(a, bb, acc);
        }
        int nn = e0 + (ln & 15);
        int rb = d0 + ((ln >> 4) << 3);
#pragma unroll
        for (int r = 0; r < 8; r++) {
          int idx = (rb + r) * DHH + nn;
          Cs[idx] = Cs[idx] * a_s + acc[r] * b_s;
        }
      }
    }
    if (tid < DHH) {
      float ksum = 0.f;
      for (int c = 0; c < CSZ; c++) ksum += ks[c * DHH + tid] * gex[c];
      ns[tid] = ns[tid] * a_s + ksum * b_s;
    }
    __syncthreads();
    if (tid == 0) m_state = m_new_s;
    __syncthreads();
  }

  // ---- final carry: Cl (B,NH,DH,DH), nl (B,NH,DH), ml (B,NH,1,1) ----
  float* Clp = out + (size_t)BB * SS * DIMM + (size_t)(b * NHH + h) * DHH * DHH;
  for (int i = tid; i < DHH * DHH; i += 256) Clp[i] = Cs[i];
  float* nlp = out + (size_t)BB * SS * DIMM + (size_t)BB * NHH * DHH * DHH
             + (size_t)(b * NHH + h) * DHH;
  if (tid < DHH) nlp[tid] = ns[tid];
  if (tid == 0)
    out[(size_t)BB * SS * DIMM + (size_t)BB * NHH * DHH * DHH + (size_t)BB * NHH * DHH
        + (b * NHH + h)] = m_state;
}

extern "C" void kernel_launch(void* const* d_in, const int* in_sizes, int n_in,
                              void* d_out, int out_size, void* d_ws, size_t ws_size,
                              hipStream_t stream) {
  const float* q   = (const float*)d_in[0];
  const float* k   = (const float*)d_in[1];
  const float* v   = (const float*)d_in[2];
  const float* Wi  = (const float*)d_in[3];
  const float* bi  = (const float*)d_in[4];
  const float* Wf  = (const float*)d_in[5];
  const float* bf  = (const float*)d_in[6];
  const float* lnw = (const float*)d_in[7];
  const float* lnb = (const float*)d_in[8];
  float* out = (float*)d_out;

  float* igw = (float*)d_ws;                      // B*NH*S floats
  float* lfw = igw + (size_t)BB * NHH * SS;       // B*NH*S floats

  mlstm_gate_kernel<<<BB * NSS, 256, 0, stream>>>(q, k, v, Wi, bi, Wf, bf, igw, lfw);
  mlstm_chunk_kernel<<<BB * NHH, 256, 0, stream>>>(q, k, v, igw, lfw, lnw, lnb, out);
}